// LinkPredictionModel_GAT_12326556140002
// MI455X (gfx1250) — compile-verified
//
#include <hip/hip_runtime.h>
#include <cmath>

#define NN 10000
#define EE 640000
#define EL (EE + NN)        // edges incl. self loops
#define HEADS 8
#define HID 16
#define FDIM 128            // HEADS*HID == IN_F == 128
#define NEG_SLOPE 0.2f

typedef __attribute__((ext_vector_type(2))) float v2f;
typedef __attribute__((ext_vector_type(8))) float v8f;

// ---------------------------------------------------------------- utilities
__global__ void fill_kernel(float* __restrict__ p, int n, float v) {
    int i = blockIdx.x * blockDim.x + threadIdx.x;
    if (i < n) p[i] = v;
}

__device__ __forceinline__ void edge_sd(const int* __restrict__ ei, int e, int& s, int& d) {
    if (e < EE) { s = ei[e]; d = ei[EE + e]; }
    else        { s = e - EE; d = s; }           // self loop
}

__device__ __forceinline__ float edge_score(const float* __restrict__ ssrc,
                                            const float* __restrict__ sdst,
                                            int s, int d, int hh) {
    float v = ssrc[s * HEADS + hh] + sdst[d * HEADS + hh];
    return v >= 0.f ? v : NEG_SLOPE * v;         // LeakyReLU(0.2)
}

// sign-aware float atomic max (memory must be initialized to -inf)
__device__ __forceinline__ void atomicMaxF(float* addr, float v) {
    if (v > 0.f) atomicMax((int*)addr, __float_as_int(v));
    else         atomicMin((unsigned int*)addr, __float_as_uint(v));
}

// ------------------------------------------------- WMMA GEMM  C[M,128] = A[M,128] * B[128,128]
// block: 256 threads = 8 waves; block handles 16 rows, each wave one 16x16 tile.
__global__ __launch_bounds__(256) void gemm128_wmma(const float* __restrict__ A,
                                                    const float* __restrict__ B,
                                                    float* __restrict__ C) {
    __shared__ __align__(16) float As[16 * FDIM];   // 8 KB A tile in LDS
    const int tid = threadIdx.x;
    const int tm  = blockIdx.x;

    // cooperative A-tile load: 2048 floats = 512 float4, 2 per thread
    const float4* A4 = (const float4*)(A + (size_t)tm * 16 * FDIM);
    float4* S4 = (float4*)As;
    S4[tid]       = A4[tid];
    S4[tid + 256] = A4[tid + 256];
    __syncthreads();

    const int wv   = tid >> 5;          // wave id -> column tile 0..7
    const int lane = tid & 31;
    const int hf   = lane >> 4;         // lane half selects K pair (ISA 7.12.2)
    const int l15  = lane & 15;
    const int col  = wv * 16 + l15;

    v8f c = {0.f, 0.f, 0.f, 0.f, 0.f, 0.f, 0.f, 0.f};
    #pragma unroll
    for (int kb = 0; kb < 32; ++kb) {   // K = 128 = 32 * 4
        const int k0 = kb * 4 + hf * 2;
        v2f a = *(const v2f*)(&As[l15 * FDIM + k0]);       // A 16x4 fragment
        v2f b;
        b.x = B[(size_t)k0 * FDIM + col];                  // B 4x16 fragment
        b.y = B[(size_t)(k0 + 1) * FDIM + col];
        c = __builtin_amdgcn_wmma_f32_16x16x4_f32(false, a, false, b,
                                                  (short)0, c, false, false);
    }
    const int rbase = tm * 16 + hf * 8;                    // C/D layout: M=i / M=8+i
    #pragma unroll
    for (int i = 0; i < 8; ++i)
        C[(size_t)(rbase + i) * FDIM + col] = c[i];
}

// ---------------------------------------------- per-node attention dot products
__global__ void attn_scores(const float* __restrict__ h,
                            const float* __restrict__ a_src,
                            const float* __restrict__ a_dst,
                            float* __restrict__ ssrc, float* __restrict__ sdst) {
    int idx = blockIdx.x * blockDim.x + threadIdx.x;      // n*HEADS + h
    if (idx >= NN * HEADS) return;
    int hh = idx & (HEADS - 1);
    int n  = idx >> 3;
    const float* hp = h + (size_t)n * FDIM + hh * HID;
    const float* as = a_src + hh * HID;
    const float* ad = a_dst + hh * HID;
    float s1 = 0.f, s2 = 0.f;
    #pragma unroll
    for (int f = 0; f < HID; ++f) { s1 += hp[f] * as[f]; s2 += hp[f] * ad[f]; }
    ssrc[idx] = s1;
    sdst[idx] = s2;
}

// ------------------------------------------------------------- edge passes
__global__ void edge_max(const int* __restrict__ ei,
                         const float* __restrict__ ssrc, const float* __restrict__ sdst,
                         float* __restrict__ m) {
    int idx = blockIdx.x * blockDim.x + threadIdx.x;      // e*HEADS + h
    if (idx >= EL * HEADS) return;
    int hh = idx & (HEADS - 1);
    int e  = idx >> 3;
    int s, d; edge_sd(ei, e, s, d);
    atomicMaxF(&m[d * HEADS + hh], edge_score(ssrc, sdst, s, d, hh));
}

__global__ void edge_denom(const int* __restrict__ ei,
                           const float* __restrict__ ssrc, const float* __restrict__ sdst,
                           const float* __restrict__ m, float* __restrict__ den) {
    int idx = blockIdx.x * blockDim.x + threadIdx.x;
    if (idx >= EL * HEADS) return;
    int hh = idx & (HEADS - 1);
    int e  = idx >> 3;
    int s, d; edge_sd(ei, e, s, d);
    float ex = expf(edge_score(ssrc, sdst, s, d, hh) - m[d * HEADS + hh]);
    atomicAdd(&den[d * HEADS + hh], ex);
}

__global__ void edge_aggregate(const int* __restrict__ ei,
                               const float* __restrict__ ssrc, const float* __restrict__ sdst,
                               const float* __restrict__ m, const float* __restrict__ den,
                               const float* __restrict__ hfeat, float* __restrict__ agg) {
    int idx = blockIdx.x * blockDim.x + threadIdx.x;
    if (idx >= EL * HEADS) return;
    int hh = idx & (HEADS - 1);
    int e  = idx >> 3;
    int s, d; edge_sd(ei, e, s, d);
    float w = expf(edge_score(ssrc, sdst, s, d, hh) - m[d * HEADS + hh])
              / den[d * HEADS + hh];
    const float* hs = hfeat + (size_t)s * FDIM + hh * HID;
    float*       od = agg   + (size_t)d * FDIM + hh * HID;
    #pragma unroll
    for (int f = 0; f < HID; ++f)
        atomicAdd(&od[f], w * hs[f]);
}

// ----------------------------------------------- epilogues
__global__ void bias_relu(float* __restrict__ a, const float* __restrict__ bias) {
    int idx = blockIdx.x * blockDim.x + threadIdx.x;
    if (idx >= NN * FDIM) return;
    a[idx] = fmaxf(a[idx] + bias[idx & (FDIM - 1)], 0.f);
}

__global__ void head_mean(const float* __restrict__ agg, const float* __restrict__ bias,
                          float* __restrict__ out) {
    int idx = blockIdx.x * blockDim.x + threadIdx.x;      // n*HID + f
    if (idx >= NN * HID) return;
    int f = idx & (HID - 1);
    int n = idx >> 4;
    const float* p = agg + (size_t)n * FDIM + f;
    float s = 0.f;
    #pragma unroll
    for (int hh = 0; hh < HEADS; ++hh) s += p[hh * HID];
    out[idx] = s * (1.f / HEADS) + bias[f];
}

// ------------------------------------------------------------------ launcher
extern "C" void kernel_launch(void* const* d_in, const int* in_sizes, int n_in,
                              void* d_out, int out_size, void* d_ws, size_t ws_size,
                              hipStream_t stream) {
    const float* x   = (const float*)d_in[0];
    const int*   ei  = (const int*)  d_in[1];
    const float* W1  = (const float*)d_in[2];
    const float* as1 = (const float*)d_in[3];
    const float* ad1 = (const float*)d_in[4];
    const float* b1  = (const float*)d_in[5];
    const float* W2  = (const float*)d_in[6];
    const float* as2 = (const float*)d_in[7];
    const float* ad2 = (const float*)d_in[8];
    const float* b2  = (const float*)d_in[9];
    float* out = (float*)d_out;

    float* w = (float*)d_ws;
    size_t o = 0;
    float* h1   = w + o; o += (size_t)NN * FDIM;
    float* agg1 = w + o; o += (size_t)NN * FDIM;   // becomes relu(h1_agg + b1)
    float* h2   = w + o; o += (size_t)NN * FDIM;
    float* agg2 = w + o; o += (size_t)NN * FDIM;
    float* ss1  = w + o; o += (size_t)NN * HEADS;
    float* sd1  = w + o; o += (size_t)NN * HEADS;
    float* m1   = w + o; o += (size_t)NN * HEADS;
    float* dn1  = w + o; o += (size_t)NN * HEADS;
    float* ss2  = w + o; o += (size_t)NN * HEADS;
    float* sd2  = w + o; o += (size_t)NN * HEADS;
    float* m2   = w + o; o += (size_t)NN * HEADS;
    float* dn2  = w + o; o += (size_t)NN * HEADS;

    const int TB = 256;
    const int gNH  = (NN * HEADS + TB - 1) / TB;
    const int gNF  = (NN * FDIM + TB - 1) / TB;
    const int gEH  = (EL * HEADS + TB - 1) / TB;
    const int gOUT = (NN * HID + TB - 1) / TB;
    const int gGEMM = NN / 16;                    // 10000 / 16 = 625 exactly

    // init (must happen every call: harness does not re-poison between replays)
    fill_kernel<<<gNH, TB, 0, stream>>>(m1,  NN * HEADS, -INFINITY);
    fill_kernel<<<gNH, TB, 0, stream>>>(dn1, NN * HEADS, 0.f);
    fill_kernel<<<gNF, TB, 0, stream>>>(agg1, NN * FDIM, 0.f);
    fill_kernel<<<gNH, TB, 0, stream>>>(m2,  NN * HEADS, -INFINITY);
    fill_kernel<<<gNH, TB, 0, stream>>>(dn2, NN * HEADS, 0.f);
    fill_kernel<<<gNF, TB, 0, stream>>>(agg2, NN * FDIM, 0.f);

    // ---- layer 1 ----
    gemm128_wmma<<<gGEMM, TB, 0, stream>>>(x, W1, h1);
    attn_scores<<<gNH, TB, 0, stream>>>(h1, as1, ad1, ss1, sd1);
    edge_max<<<gEH, TB, 0, stream>>>(ei, ss1, sd1, m1);
    edge_denom<<<gEH, TB, 0, stream>>>(ei, ss1, sd1, m1, dn1);
    edge_aggregate<<<gEH, TB, 0, stream>>>(ei, ss1, sd1, m1, dn1, h1, agg1);
    bias_relu<<<gNF, TB, 0, stream>>>(agg1, b1);

    // ---- layer 2 ----
    gemm128_wmma<<<gGEMM, TB, 0, stream>>>(agg1, W2, h2);
    attn_scores<<<gNH, TB, 0, stream>>>(h2, as2, ad2, ss2, sd2);
    edge_max<<<gEH, TB, 0, stream>>>(ei, ss2, sd2, m2);
    edge_denom<<<gEH, TB, 0, stream>>>(ei, ss2, sd2, m2, dn2);
    edge_aggregate<<<gEH, TB, 0, stream>>>(ei, ss2, sd2, m2, dn2, h2, agg2);
    head_mean<<<gOUT, TB, 0, stream>>>(agg2, b2, out);
}